// PointMonarch_38216619000501
// MI455X (gfx1250) — compile-verified
//
#include <hip/hip_runtime.h>
#include <hip/hip_bf16.h>

typedef float v2f __attribute__((ext_vector_type(2)));
typedef float v8f __attribute__((ext_vector_type(8)));
typedef int   v4i __attribute__((vector_size(16)));   // matches builtin param type

#define AS1 __attribute__((address_space(1)))
#define AS3 __attribute__((address_space(3)))

#if __has_builtin(__builtin_amdgcn_global_load_async_to_lds_b128) && \
    __has_builtin(__builtin_amdgcn_s_wait_asynccnt)
#define PM_ASYNC_WLOAD 1
#else
#define PM_ASYNC_WLOAD 0
#endif

namespace {
constexpr int   Bb    = 8;      // batch
constexpr int   Nn    = 8192;   // points
constexpr int   Cc    = 256;    // channels
constexpr int   Kk    = 8;      // neighbors
constexpr int   NBLK  = 8;      // monarch blocks
constexpr int   BLKd  = 32;     // block dim (32x32)
constexpr int   ROWS  = 16;     // rows per tile (= one WMMA M tile)
constexpr int   TILES = 8;      // row-tiles per workgroup (amortizes weight load)
constexpr float EPS   = 0.05f;
constexpr int   S1S   = 260;    // s1/o1 row stride: 260%64==4 -> conflict-free row-strided reads
constexpr int   WST   = 36;     // weight row stride: 144B rows (16B-aligned chunks), 36q%64 distinct
constexpr int   O2S   = 292;    // o2 staging stride: +8 rows lands 32 banks away
}

// Per-row softmax(-distance) * EPS into wt[ROWS*K], neighbor indices into nb[ROWS*K].
__device__ __forceinline__ void load_wt_nb(const float* __restrict__ dist,
                                           const int* __restrict__ idx,
                                           int b, int n0, float* wt, int* nb, int tid) {
    if (tid < ROWS * Kk)
        nb[tid] = idx[((size_t)b * Nn + n0 + (tid >> 3)) * Kk + (tid & 7)];
    if (tid < ROWS) {
        const float* dp = dist + ((size_t)b * Nn + n0 + tid) * Kk;
        float d0[Kk];
        float mn = 1e30f;
#pragma unroll
        for (int k = 0; k < Kk; ++k) { d0[k] = dp[k]; mn = fminf(mn, d0[k]); }
        float s = 0.f, e0[Kk];
#pragma unroll
        for (int k = 0; k < Kk; ++k) { e0[k] = __expf(mn - d0[k]); s += e0[k]; }
        float inv = EPS / s;   // fold eps into the weights
#pragma unroll
        for (int k = 0; k < Kk; ++k) wt[tid * Kk + k] = e0[k] * inv;
    }
}

// Kernel 1: rectify1 (gather-blend) fused with monarch (two WMMA f32 16x16x4 GEMM stages).
// Each block: 8 row-tiles of 16 rows. 8 waves; wave w owns monarch block w.
// Weights are async-copied to LDS once per block (overlapped with tile-0 rectify),
// then each wave's B-fragments live in registers for the whole tile loop.
__global__ __launch_bounds__(256) void pm_rect1_monarch(
        const float* __restrict__ x, const float* __restrict__ dist,
        const int*  __restrict__ idx, const float* __restrict__ w1,
        const float* __restrict__ w2, const float* __restrict__ bias,
        float* __restrict__ t)
{
    __shared__ float w1l[NBLK * BLKd * WST];   // w1[k][q][p] padded
    __shared__ float w2l[NBLK * BLKd * WST];   // w2[l][s][r] padded
    __shared__ float s1 [ROWS * S1S];          // rectified tile
    __shared__ float o1 [ROWS * S1S];          // stage-1 output (pre-butterfly)
    __shared__ float o2s[ROWS * O2S];          // stage-2 staging: [row][l*WST + s]
    __shared__ float wt [ROWS * Kk];
    __shared__ int   nb [ROWS * Kk];

    const int tid   = threadIdx.x;
    const int lane  = tid & 31;
    const int wave  = tid >> 5;
    const int l16   = lane & 15;
    const int khalf = (lane >> 4) << 1;        // K half select: 0 / 2
    const int mhalf = (lane >> 4) << 3;        // D-matrix M half: 0 / 8
    const int tile0 = blockIdx.x * TILES;

#if PM_ASYNC_WLOAD
    // Issue async global->LDS weight copy (16B per lane per op); completes before stage 1.
#pragma unroll
    for (int it = 0; it < 8; ++it) {
        const int i   = it * 256 + tid;        // 16B chunk id, 2048 per weight array
        const int row = i >> 3;                // padded row (k*32+q)
        const int off = (i & 7) << 2;          // float offset within row
        __builtin_amdgcn_global_load_async_to_lds_b128(
            (AS1 v4i*)(w1 + 4 * (size_t)i),
            (AS3 v4i*)&w1l[row * WST + off], 0, 0);
        __builtin_amdgcn_global_load_async_to_lds_b128(
            (AS1 v4i*)(w2 + 4 * (size_t)i),
            (AS3 v4i*)&w2l[row * WST + off], 0, 0);
    }
#else
    for (int i = tid; i < NBLK * BLKd * BLKd; i += 256) {
        const int kq = i >> 5, p = i & 31;
        w1l[kq * WST + p] = w1[i];
        w2l[kq * WST + p] = w2[i];
    }
#endif

    const float bv = bias[tid];
    const int   lo = tid & 7;                  // output un-permute: c = s*8 + l
    const int   so = tid >> 3;

    v2f bf1[16];                               // stage-1 B frags [j*8+kk], resident in VGPRs
    v2f bf2[16];                               // stage-2 B frags

    for (int ti = 0; ti < TILES; ++ti) {
        const int tile = tile0 + ti;
        const int b    = tile >> 9;            // 512 tiles per batch
        const int n0   = (tile & 511) * ROWS;

        load_wt_nb(dist, idx, b, n0, wt, nb, tid);
        __syncthreads();                       // wt/nb ready; prior-tile LDS readers done

        // ---- Phase 1: distance-weighted neighbor smoothing into s1 (tid == channel) ----
        const float* xb = x + (size_t)b * Nn * Cc;
#pragma unroll 4
        for (int r = 0; r < ROWS; ++r) {
            float acc = (1.0f - EPS) * xb[(size_t)(n0 + r) * Cc + tid];
#pragma unroll
            for (int k = 0; k < Kk; ++k)
                acc += wt[r * Kk + k] * xb[(size_t)nb[r * Kk + k] * Cc + tid];
            s1[r * S1S + tid] = acc;
        }
#if PM_ASYNC_WLOAD
        if (ti == 0) __builtin_amdgcn_s_wait_asynccnt(0);
#endif
        __syncthreads();                       // s1 ready (+ weights on first tile)

        if (ti == 0) {                         // one-time B-fragment hoist into registers
#pragma unroll
            for (int j = 0; j < 2; ++j)
#pragma unroll
                for (int kk = 0; kk < 8; ++kk) {
                    const int qrow = (wave * BLKd + j * 16 + l16) * WST + kk * 4 + khalf;
                    bf1[j * 8 + kk] = *(const v2f*)&w1l[qrow];
                    bf2[j * 8 + kk] = *(const v2f*)&w2l[qrow];
                }
        }

        // ---- Stage 1: o1[r][k*32+q] = sum_p w1[k][q][p] * s1[r][k*32+p] ----
#pragma unroll
        for (int j = 0; j < 2; ++j) {
            v8f acc = {};
#pragma unroll
            for (int kk = 0; kk < 8; ++kk) {
                const int p0 = kk * 4 + khalf;
                v2f a = *(const v2f*)&s1[l16 * S1S + wave * BLKd + p0];   // A[m][K]
                acc = __builtin_amdgcn_wmma_f32_16x16x4_f32(
                        false, a, false, bf1[j * 8 + kk], (short)0, acc, false, false);
            }
#pragma unroll
            for (int v = 0; v < 8; ++v)        // D: vgpr v -> row (v+mhalf), col q
                o1[(v + mhalf) * S1S + wave * BLKd + j * 16 + l16] = acc[v];
        }
        __syncthreads();

        // ---- Stage 2: butterfly (stride-8 gather) + o2[r][l][s] = sum w2[l][s][.]*o1flat ----
#pragma unroll
        for (int j = 0; j < 2; ++j) {
            v8f acc = {};
#pragma unroll
            for (int kk = 0; kk < 8; ++kk) {
                const int r0 = kk * 4 + khalf;
                v2f a;                         // A[m][K=r] = o1[m][r*8 + wave]
                a.x = o1[l16 * S1S + r0 * NBLK + wave];
                a.y = o1[l16 * S1S + (r0 + 1) * NBLK + wave];
                acc = __builtin_amdgcn_wmma_f32_16x16x4_f32(
                        false, a, false, bf2[j * 8 + kk], (short)0, acc, false, false);
            }
#pragma unroll
            for (int v = 0; v < 8; ++v)
                o2s[(v + mhalf) * O2S + wave * WST + j * 16 + l16] = acc[v];
        }
        __syncthreads();

        // ---- Un-permute (c = s*8 + l), add bias, coalesced store of t ----
        float* tb = t + (size_t)b * Nn * Cc;
#pragma unroll 4
        for (int r = 0; r < ROWS; ++r)
            tb[(size_t)(n0 + r) * Cc + tid] = o2s[r * O2S + lo * WST + so] + bv;
    }
}

// Kernel 2: rectify2 on t + residual rf. Fully coalesced: all 256 lanes walk one row together.
__global__ __launch_bounds__(256) void pm_rect2_residual(
        const float* __restrict__ t, const float* __restrict__ dist,
        const int*  __restrict__ idx, const float* __restrict__ rf,
        float* __restrict__ out)
{
    __shared__ float wt[ROWS * Kk];
    __shared__ int   nb[ROWS * Kk];

    const int tid = threadIdx.x;
    const int b   = blockIdx.x / (Nn / ROWS);
    const int n0  = (blockIdx.x % (Nn / ROWS)) * ROWS;

    load_wt_nb(dist, idx, b, n0, wt, nb, tid);
    __syncthreads();

    const float* tb = t + (size_t)b * Nn * Cc;
    for (int r = 0; r < ROWS; ++r) {
        // Prefetch next row's (random) neighbor rows while this row computes.
        if (r + 1 < ROWS && tid < Kk)
            __builtin_prefetch(&tb[(size_t)nb[(r + 1) * Kk + tid] * Cc], 0, 0);

        const size_t o = ((size_t)b * Nn + n0 + r) * Cc + tid;
        float acc = (1.0f - EPS) * tb[(size_t)(n0 + r) * Cc + tid];
#pragma unroll
        for (int k = 0; k < Kk; ++k)
            acc += wt[r * Kk + k] * tb[(size_t)nb[r * Kk + k] * Cc + tid];
        out[o] = acc + rf[o];
    }
}

extern "C" void kernel_launch(void* const* d_in, const int* in_sizes, int n_in,
                              void* d_out, int out_size, void* d_ws, size_t ws_size,
                              hipStream_t stream) {
    (void)in_sizes; (void)n_in; (void)out_size; (void)ws_size;
    const float* x    = (const float*)d_in[0];
    const float* dist = (const float*)d_in[1];
    const int*   idx  = (const int*)  d_in[2];
    const float* rf   = (const float*)d_in[3];
    const float* w1   = (const float*)d_in[4];
    const float* w2   = (const float*)d_in[5];
    const float* bias = (const float*)d_in[6];
    float*       out  = (float*)d_out;
    float*       t    = (float*)d_ws;           // needs B*N*C*4 = 64 MB scratch

    pm_rect1_monarch <<<dim3(Bb * (Nn / ROWS) / TILES), 256, 0, stream>>>(
            x, dist, idx, w1, w2, bias, t);
    pm_rect2_residual<<<dim3(Bb * (Nn / ROWS)),         256, 0, stream>>>(
            t, dist, idx, rf, out);
}